// HeteroGNN_21105469292717
// MI455X (gfx1250) — compile-verified
//
#include <hip/hip_runtime.h>
#include <hip/hip_bf16.h>

typedef __attribute__((ext_vector_type(2))) float v2f;
typedef __attribute__((ext_vector_type(8))) float v8f;

#define NEG_SLOPE 0.2f

// ---------------------------------------------------------------------------
// wd_a[i] = sum_j Wd[i][j] * a_d[j]   (collapses dst-side GEMM into a matvec)
__global__ void wv_kernel(const float* __restrict__ Wd, const float* __restrict__ a,
                          float* __restrict__ out, int C, int F) {
    int i = blockIdx.x * blockDim.x + threadIdx.x;
    if (i >= C) return;
    float acc = 0.f;
    for (int j = 0; j < F; ++j) acc += Wd[i * F + j] * a[j];
    out[i] = acc;
}

// ---------------------------------------------------------------------------
// H[n x 16] = X[n x 64] @ W[64 x 16]  via V_WMMA_F32_16X16X4_F32, one wave per
// 16-row tile, 16 K-steps. Fused epilogue: alphaS[row] = dot(H[row], a_s)
// using half-wave shfl_xor reductions (wave32; masks 1/2/4/8 stay in 16-lane
// halves, matching the C/D layout where lanes 0-15 hold M=i, 16-31 hold M=i+8).
__global__ void gemm64x16_wmma(const float* __restrict__ X, const float* __restrict__ W,
                               const float* __restrict__ a_s, float* __restrict__ H,
                               float* __restrict__ alphaS, int nTiles) {
    const int lane = threadIdx.x & 31;
    const int wave = threadIdx.x >> 5;
    const int tile = blockIdx.x * (blockDim.x >> 5) + wave;
    if (tile >= nTiles) return;

    const int half = lane >> 4;   // 0: lanes 0-15, 1: lanes 16-31
    const int l16  = lane & 15;

    v8f acc = {};
    const size_t rowA = (size_t)(tile * 16 + l16) * 64;
#pragma unroll
    for (int k0 = 0; k0 < 64; k0 += 4) {
        // A 16x4 f32: lane l16 = row M; half selects K pair {k0,k0+1} / {k0+2,k0+3}
        float2 av = *(const float2*)(X + rowA + k0 + 2 * half);
        // B 4x16 f32: lane l16 = col N; half selects same K pair
        v2f A, B;
        A.x = av.x; A.y = av.y;
        B.x = W[(k0 + 2 * half + 0) * 16 + l16];
        B.y = W[(k0 + 2 * half + 1) * 16 + l16];
        acc = __builtin_amdgcn_wmma_f32_16x16x4_f32(
            false, A, false, B, (short)0, acc, false, false);
    }

    const float aw = a_s[l16];
#pragma unroll
    for (int i = 0; i < 8; ++i) {
        int row = tile * 16 + half * 8 + i;   // C/D layout: VGPR i -> M=i (+8 for hi half)
        float c = acc[i];
        H[(size_t)row * 16 + l16] = c;
        float v = c * aw;                      // partial of dot(H[row], a_s)
        v += __shfl_xor(v, 1);
        v += __shfl_xor(v, 2);
        v += __shfl_xor(v, 4);
        v += __shfl_xor(v, 8);
        if (l16 == 0) alphaS[row] = v;
    }
}

// ---------------------------------------------------------------------------
// out[i] = dot(X[i, 0:C], w[0:C])   (C multiple of 4)
__global__ void rowdot_kernel(const float* __restrict__ X, const float* __restrict__ w,
                              float* __restrict__ out, int n, int C) {
    int i = blockIdx.x * blockDim.x + threadIdx.x;
    if (i >= n) return;
    const float4* xr = (const float4*)(X + (size_t)i * C);
    float acc = 0.f;
    for (int j = 0; j < (C >> 2); ++j) {
        float4 x = xr[j];
        const float4 ww = ((const float4*)w)[j];
        acc += x.x * ww.x + x.y * ww.y + x.z * ww.z + x.w * ww.w;
    }
    out[i] = acc;
}

// ---------------------------------------------------------------------------
// per-dst init: segment-max key = INT_MIN, segment-sum = 0, out row = bias
__global__ void init_dst_kernel(int* __restrict__ m, float* __restrict__ s,
                                float* __restrict__ out, const float* __restrict__ b,
                                int n, int F) {
    int i = blockIdx.x * blockDim.x + threadIdx.x;
    if (i >= n) return;
    m[i] = (int)0x80000000;
    s[i] = 0.f;
    for (int c = 0; c < F; ++c) out[(size_t)i * F + c] = b[c];
}

__device__ __forceinline__ int f2key(float f) {
    int b = __float_as_int(f);
    return b >= 0 ? b : (b ^ 0x7fffffff);   // monotone float->int ordering
}
__device__ __forceinline__ float key2f(int k) {
    return __int_as_float(k >= 0 ? k : (k ^ 0x7fffffff));
}

// pass 1: e = leaky_relu(aS[src]+aD[dst]); store e; segment max via atomicMax
__global__ void edge_max_kernel(const int* __restrict__ src, const int* __restrict__ dst,
                                const float* __restrict__ aS, const float* __restrict__ aD,
                                float* __restrict__ ebuf, int* __restrict__ m, int E) {
    int k = blockIdx.x * blockDim.x + threadIdx.x;
    if (k >= E) return;
    int d = dst[k];
    float e = aS[src[k]] + aD[d];
    e = e >= 0.f ? e : NEG_SLOPE * e;
    ebuf[k] = e;
    atomicMax(&m[d], f2key(e));
}

// pass 2: w = exp(e - m[dst]); store w; segment sum via atomicAdd
__global__ void edge_exp_kernel(const int* __restrict__ dst, float* __restrict__ ebuf,
                                const int* __restrict__ m, float* __restrict__ s, int E) {
    int k = blockIdx.x * blockDim.x + threadIdx.x;
    if (k >= E) return;
    int d = dst[k];
    float w = __expf(ebuf[k] - key2f(m[d]));
    ebuf[k] = w;
    atomicAdd(&s[d], w);
}

// pass 3 (F=16): out[dst] += (w / (s[dst]+eps)) * H[src]
__global__ void edge_agg16_kernel(const int* __restrict__ src, const int* __restrict__ dst,
                                  const float* __restrict__ ebuf, const float* __restrict__ s,
                                  const float* __restrict__ H, float* __restrict__ out, int E) {
    int k = blockIdx.x * blockDim.x + threadIdx.x;
    if (k >= E) return;
    int d = dst[k], sr = src[k];
    float alpha = ebuf[k] / (s[d] + 1e-16f);
    const float4* h = (const float4*)(H + (size_t)sr * 16);
    float* o = out + (size_t)d * 16;
#pragma unroll
    for (int q = 0; q < 4; ++q) {
        float4 hv = h[q];
        atomicAdd(o + q * 4 + 0, alpha * hv.x);
        atomicAdd(o + q * 4 + 1, alpha * hv.y);
        atomicAdd(o + q * 4 + 2, alpha * hv.z);
        atomicAdd(o + q * 4 + 3, alpha * hv.w);
    }
}

// pass 3 (F=2)
__global__ void edge_agg2_kernel(const int* __restrict__ src, const int* __restrict__ dst,
                                 const float* __restrict__ ebuf, const float* __restrict__ s,
                                 const float* __restrict__ H, float* __restrict__ out, int E) {
    int k = blockIdx.x * blockDim.x + threadIdx.x;
    if (k >= E) return;
    int d = dst[k], sr = src[k];
    float alpha = ebuf[k] / (s[d] + 1e-16f);
    float2 hv = *(const float2*)(H + (size_t)sr * 2);
    atomicAdd(out + (size_t)d * 2 + 0, alpha * hv.x);
    atomicAdd(out + (size_t)d * 2 + 1, alpha * hv.y);
}

// layer-2 src projection: H2[n x 2] = X[n x 16] @ W[16 x 2]; alphaS = H2 . a
__global__ void gemm16x2_kernel(const float* __restrict__ X, const float* __restrict__ W,
                                const float* __restrict__ a, float* __restrict__ H2,
                                float* __restrict__ alphaS, int n) {
    int i = blockIdx.x * blockDim.x + threadIdx.x;
    if (i >= n) return;
    const float4* xr = (const float4*)(X + (size_t)i * 16);
    float h0 = 0.f, h1 = 0.f;
#pragma unroll
    for (int q = 0; q < 4; ++q) {
        float4 x = xr[q];
        h0 += x.x * W[(q * 4 + 0) * 2 + 0] + x.y * W[(q * 4 + 1) * 2 + 0]
            + x.z * W[(q * 4 + 2) * 2 + 0] + x.w * W[(q * 4 + 3) * 2 + 0];
        h1 += x.x * W[(q * 4 + 0) * 2 + 1] + x.y * W[(q * 4 + 1) * 2 + 1]
            + x.z * W[(q * 4 + 2) * 2 + 1] + x.w * W[(q * 4 + 3) * 2 + 1];
    }
    H2[(size_t)i * 2 + 0] = h0;
    H2[(size_t)i * 2 + 1] = h1;
    alphaS[i] = h0 * a[0] + h1 * a[1];
}

__global__ void relu_kernel(float* __restrict__ x, int n) {
    int i = blockIdx.x * blockDim.x + threadIdx.x;
    if (i >= n) return;
    x[i] = fmaxf(x[i], 0.f);
}

// ---------------------------------------------------------------------------
static inline int cdiv(int a, int b) { return (a + b - 1) / b; }

extern "C" void kernel_launch(void* const* d_in, const int* in_sizes, int n_in,
                              void* d_out, int out_size, void* d_ws, size_t ws_size,
                              hipStream_t stream) {
    (void)in_sizes; (void)n_in; (void)out_size; (void)ws_size;
    const int NU = 200000, NB = 50000, E = 5000000;

    const float* x_user  = (const float*)d_in[0];
    const float* x_badge = (const float*)d_in[1];
    const int* ub_src = (const int*)d_in[2];
    const int* ub_dst = (const int*)d_in[3];
    const int* bu_src = (const int*)d_in[4];
    const int* bu_dst = (const int*)d_in[5];
    // d_in[6], d_in[7]: edge weights — unused (edge_dim=None in reference)
    const float* W1_ub_s = (const float*)d_in[8];
    const float* W1_ub_d = (const float*)d_in[9];
    const float* a1_ub_s = (const float*)d_in[10];
    const float* a1_ub_d = (const float*)d_in[11];
    const float* b1_ub   = (const float*)d_in[12];
    const float* W2_ub_s = (const float*)d_in[13];
    const float* W2_ub_d = (const float*)d_in[14];
    const float* a2_ub_s = (const float*)d_in[15];
    const float* a2_ub_d = (const float*)d_in[16];
    const float* b2_ub   = (const float*)d_in[17];
    const float* W1_bu_s = (const float*)d_in[18];
    const float* W1_bu_d = (const float*)d_in[19];
    const float* a1_bu_s = (const float*)d_in[20];
    const float* a1_bu_d = (const float*)d_in[21];
    const float* b1_bu   = (const float*)d_in[22];
    const float* W2_bu_s = (const float*)d_in[23];
    const float* W2_bu_d = (const float*)d_in[24];
    const float* a2_bu_s = (const float*)d_in[25];
    const float* a2_bu_d = (const float*)d_in[26];
    const float* b2_bu   = (const float*)d_in[27];

    float* out = (float*)d_out;            // [ou: 200000*2][ob: 50000*2]
    float* ou  = out;
    float* ob  = out + (size_t)NU * 2;

    // workspace carve-out (~52 MB of fp32)
    float* ws      = (float*)d_ws;
    float* hu_buf  = ws;                        // 200000*16
    float* hb_buf  = hu_buf + (size_t)NU * 16;  // 50000*16
    float* hs_buf  = hb_buf + (size_t)NB * 16;  // reused per relation (<= 200000*16)
    float* alphaS  = hs_buf + (size_t)NU * 16;  // 200000
    float* alphaD  = alphaS + NU;               // 200000
    float* s_buf   = alphaD + NU;               // 200000
    int*   m_buf   = (int*)(s_buf + NU);        // 200000
    float* e_buf   = (float*)(m_buf + NU);      // 5000000
    float* wdA1_ub = e_buf + E;                 // 64
    float* wdA1_bu = wdA1_ub + 64;              // 64
    float* wdA2_ub = wdA1_bu + 64;              // 16
    float* wdA2_bu = wdA2_ub + 16;              // 16

    const int T = 256;
    const int egrid = cdiv(E, T);

    // fold dst-side projections: wd_a = Wd @ a_d
    wv_kernel<<<1, 64, 0, stream>>>(W1_ub_d, a1_ub_d, wdA1_ub, 64, 16);
    wv_kernel<<<1, 64, 0, stream>>>(W1_bu_d, a1_bu_d, wdA1_bu, 64, 16);
    wv_kernel<<<1, 16, 0, stream>>>(W2_ub_d, a2_ub_d, wdA2_ub, 16, 2);
    wv_kernel<<<1, 16, 0, stream>>>(W2_bu_d, a2_bu_d, wdA2_bu, 16, 2);

    // ---- Layer 1, relation ub: src=user, dst=badge -> hb_buf
    gemm64x16_wmma<<<cdiv(NU / 16, 8), T, 0, stream>>>(x_user, W1_ub_s, a1_ub_s, hs_buf, alphaS, NU / 16);
    rowdot_kernel<<<cdiv(NB, T), T, 0, stream>>>(x_badge, wdA1_ub, alphaD, NB, 64);
    init_dst_kernel<<<cdiv(NB, T), T, 0, stream>>>(m_buf, s_buf, hb_buf, b1_ub, NB, 16);
    edge_max_kernel<<<egrid, T, 0, stream>>>(ub_src, ub_dst, alphaS, alphaD, e_buf, m_buf, E);
    edge_exp_kernel<<<egrid, T, 0, stream>>>(ub_dst, e_buf, m_buf, s_buf, E);
    edge_agg16_kernel<<<egrid, T, 0, stream>>>(ub_src, ub_dst, e_buf, s_buf, hs_buf, hb_buf, E);

    // ---- Layer 1, relation bu: src=badge, dst=user -> hu_buf
    gemm64x16_wmma<<<cdiv(NB / 16, 8), T, 0, stream>>>(x_badge, W1_bu_s, a1_bu_s, hs_buf, alphaS, NB / 16);
    rowdot_kernel<<<cdiv(NU, T), T, 0, stream>>>(x_user, wdA1_bu, alphaD, NU, 64);
    init_dst_kernel<<<cdiv(NU, T), T, 0, stream>>>(m_buf, s_buf, hu_buf, b1_bu, NU, 16);
    edge_max_kernel<<<egrid, T, 0, stream>>>(bu_src, bu_dst, alphaS, alphaD, e_buf, m_buf, E);
    edge_exp_kernel<<<egrid, T, 0, stream>>>(bu_dst, e_buf, m_buf, s_buf, E);
    edge_agg16_kernel<<<egrid, T, 0, stream>>>(bu_src, bu_dst, e_buf, s_buf, hs_buf, hu_buf, E);

    relu_kernel<<<cdiv(NU * 16, T), T, 0, stream>>>(hu_buf, NU * 16);
    relu_kernel<<<cdiv(NB * 16, T), T, 0, stream>>>(hb_buf, NB * 16);

    // ---- Layer 2, relation ub: src=hu, dst=hb -> ob
    gemm16x2_kernel<<<cdiv(NU, T), T, 0, stream>>>(hu_buf, W2_ub_s, a2_ub_s, hs_buf, alphaS, NU);
    rowdot_kernel<<<cdiv(NB, T), T, 0, stream>>>(hb_buf, wdA2_ub, alphaD, NB, 16);
    init_dst_kernel<<<cdiv(NB, T), T, 0, stream>>>(m_buf, s_buf, ob, b2_ub, NB, 2);
    edge_max_kernel<<<egrid, T, 0, stream>>>(ub_src, ub_dst, alphaS, alphaD, e_buf, m_buf, E);
    edge_exp_kernel<<<egrid, T, 0, stream>>>(ub_dst, e_buf, m_buf, s_buf, E);
    edge_agg2_kernel<<<egrid, T, 0, stream>>>(ub_src, ub_dst, e_buf, s_buf, hs_buf, ob, E);

    // ---- Layer 2, relation bu: src=hb, dst=hu -> ou
    gemm16x2_kernel<<<cdiv(NB, T), T, 0, stream>>>(hb_buf, W2_bu_s, a2_bu_s, hs_buf, alphaS, NB);
    rowdot_kernel<<<cdiv(NU, T), T, 0, stream>>>(hu_buf, wdA2_bu, alphaD, NU, 16);
    init_dst_kernel<<<cdiv(NU, T), T, 0, stream>>>(m_buf, s_buf, ou, b2_bu, NU, 2);
    edge_max_kernel<<<egrid, T, 0, stream>>>(bu_src, bu_dst, alphaS, alphaD, e_buf, m_buf, E);
    edge_exp_kernel<<<egrid, T, 0, stream>>>(bu_dst, e_buf, m_buf, s_buf, E);
    edge_agg2_kernel<<<egrid, T, 0, stream>>>(bu_src, bu_dst, e_buf, s_buf, hs_buf, ou, E);
}